// CKConv_63282048139273
// MI455X (gfx1250) — compile-verified
//
#include <hip/hip_runtime.h>

// Problem constants (reference: D=64, HID=50, SIREN omega0=30, LN eps=1e-5)
#define DIM     64
#define HID     50
#define KPAD    64                  // padded inner dim: 50 (h2) + 1 (bias row) + zeros
#define NTILES  ((DIM * DIM) / 16)  // 256 column tiles of the 4096-wide kernel GEMM
#define KSTEPS  (KPAD / 4)          // 16 f32 WMMA k-steps (K=4 each)
#define OMEGA0  30.0f
#define LNEPS   1e-5f
#define WPB     4                   // waves per block
#define BLOCK   (WPB * 32)
#define RSTRIDE 68                  // padded LDS row stride (floats) -> conflict-free

typedef float v2f __attribute__((ext_vector_type(2)));
typedef float v8f __attribute__((ext_vector_type(8)));

__global__ void zero_f32(float* __restrict__ p, int n) {
  int i = blockIdx.x * blockDim.x + threadIdx.x;
  if (i < n) p[i] = 0.f;
}

// Repack W3 (HID x D*D) with b3 folded in as K-row HID, zero-padded to KPAD,
// into WMMA B-operand lane order (mirrors the ISA A-layout K order):
//   wp[((nt*KSTEPS + ks)*32 + lane)*2 + v] = W3aug[4*ks + 2*(lane/16) + v][nt*16 + lane%16]
__global__ void pack_w3(const float* __restrict__ W3, const float* __restrict__ b3,
                        float* __restrict__ wp) {
  int flat = blockIdx.x * blockDim.x + threadIdx.x;
  if (flat >= NTILES * KSTEPS * 32) return;
  int lane = flat & 31;
  int nt   = flat >> 9;              // /(16*32)
  int n    = nt * 16 + (lane & 15);
  int kb   = 4 * ((flat >> 5) & 15) + 2 * (lane >> 4);
  float vals[2];
#pragma unroll
  for (int v = 0; v < 2; ++v) {
    int k = kb + v;
    vals[v] = (k < HID) ? W3[k * (DIM * DIM) + n] : ((k == HID) ? b3[n] : 0.f);
  }
  float2 o; o.x = vals[0]; o.y = vals[1];
  ((float2*)wp)[flat] = o;
}

// Fused: SIREN MLP (scalar t -> h2[50]) -> K = h2aug @ W3aug (f32 WMMA)
//        -> msg[e] = K[e] @ x[e] (folded per tile) -> atomic segment-sum.
// One wave owns 16 edges. A-operand (16x64) lives in 16 v2f registers and is
// reused across all 256 N-tiles; B streams from the 1MB L2-hot packed W3.
// Two independent WMMA accumulator chains per tile keep the matrix pipe busy.
__launch_bounds__(BLOCK)
__global__ void ckconv_side(
    const float* __restrict__ timeSrc, const int* __restrict__ timeIdx,
    const float* __restrict__ edgesT,
    const float* __restrict__ emb,      const int* __restrict__ gatherIdx,
    const int*   __restrict__ scatterIdx,
    const float* __restrict__ W1, const float* __restrict__ b1,
    const float* __restrict__ g1, const float* __restrict__ be1,
    const float* __restrict__ W2, const float* __restrict__ b2,
    const float* __restrict__ g2, const float* __restrict__ be2,
    const float* __restrict__ wpack,
    float* __restrict__ out, int E)
{
  __shared__ float sW2[HID * HID];
  __shared__ float sP[7 * HID];  // W1,b1,g1,be1,b2,g2,be2
  __shared__ __align__(16) float h2t[WPB][16][RSTRIDE];  // padded A tiles
  __shared__ __align__(16) float xt [WPB][16][RSTRIDE];  // gathered x tiles

  const int tid  = threadIdx.x;
  const int lane = tid & 31;
  const int wave = tid >> 5;

  for (int i = tid; i < HID * HID; i += BLOCK) sW2[i] = W2[i];
  if (tid < HID) {
    sP[0 * HID + tid] = W1[tid];  sP[1 * HID + tid] = b1[tid];
    sP[2 * HID + tid] = g1[tid];  sP[3 * HID + tid] = be1[tid];
    sP[4 * HID + tid] = b2[tid];  sP[5 * HID + tid] = g2[tid];
    sP[6 * HID + tid] = be2[tid];
  }
  __syncthreads();

  const int  nGroups    = (E + 15) >> 4;
  const int  group      = blockIdx.x * WPB + wave;
  const bool activeWave = (group < nGroups);
  const int  eb         = group << 4;

  int gidx = 0, sidx = 0;
  if (activeWave) {
    const int m0 = lane & 15;
    int e = eb + m0;
    const float valid = (e < E) ? 1.f : 0.f;  // partial-tail edges contribute 0
    if (e >= E) e = 0;
    gidx = gatherIdx[e];
    sidx = scatterIdx[e];

    // gather x rows (16 x 64 f32) into LDS with 16B vector loads
#pragma unroll
    for (int it = 0; it < 8; ++it) {
      int f = it * 32 + lane;          // 0..255 : (row, float4-chunk)
      int m = f >> 4;
      int c = f & 15;
      int row = __shfl(gidx, m, 32);
      float4 xv = *(const float4*)(emb + (long)row * DIM + c * 4);
      *(float4*)&xt[wave][m][c * 4] = xv;
    }

    // SIREN MLP: lanes 0..15, one edge each
    if (lane < 16) {
      float t = timeSrc[timeIdx[e]] - edgesT[e];
      float h[HID];
      float s0 = 0.f, s1 = 0.f;
#pragma unroll
      for (int j = 0; j < HID; ++j) {
        float v = fmaf(t, sP[0 * HID + j], sP[1 * HID + j]) * OMEGA0;
        h[j] = v; s0 += v; s1 += v * v;
      }
      float mean = s0 * (1.f / HID);
      float rinv = rsqrtf(s1 * (1.f / HID) - mean * mean + LNEPS);
#pragma unroll
      for (int j = 0; j < HID; ++j)
        h[j] = sinf(fmaf((h[j] - mean) * rinv, sP[2 * HID + j], sP[3 * HID + j]));

      // layer 2: stream pre-LN outputs through our LDS row
      s0 = 0.f; s1 = 0.f;
      for (int j = 0; j < HID; ++j) {
        float acc = sP[4 * HID + j];
#pragma unroll
        for (int k = 0; k < HID; ++k) acc = fmaf(h[k], sW2[k * HID + j], acc);
        acc *= OMEGA0;
        h2t[wave][m0][j] = acc;
        s0 += acc; s1 += acc * acc;
      }
      mean = s0 * (1.f / HID);
      rinv = rsqrtf(s1 * (1.f / HID) - mean * mean + LNEPS);
      for (int j = 0; j < HID; ++j) {
        float v = h2t[wave][m0][j];
        h2t[wave][m0][j] =
            valid * sinf(fmaf((v - mean) * rinv, sP[5 * HID + j], sP[6 * HID + j]));
      }
      h2t[wave][m0][HID] = valid;   // constant-1 column -> adds b3 row of W3aug
#pragma unroll
      for (int j = HID + 1; j < KPAD; ++j) h2t[wave][m0][j] = 0.f;
    }
  }
  __syncthreads();
  if (!activeWave) return;  // wave-uniform; no barriers after this point

  // A operand: 16 edges x 64 k in 16 v2f regs.
  // ISA layout (32-bit A 16x4): lane m=lane%16, k = 4*ks + 2*(lane/16) + v.
  const int mA  = lane & 15;
  const int hiA = lane >> 4;
  v2f a[KSTEPS];
#pragma unroll
  for (int ks = 0; ks < KSTEPS; ++ks) {
    int k = 4 * ks + 2 * hiA;
    a[ks].x = h2t[wave][mA][k];
    a[ks].y = h2t[wave][mA][k + 1];
  }

  // Scatter mapping: after butterfly, lanes 0..7 write m=lane, lanes 16..23 write m=lane-8.
  const bool wr    = (lane < 8) || (lane >= 16 && lane < 24);
  const int  myM   = ((lane < 8) ? lane : (lane - 8)) & 15;
  const int  myRow = __shfl(sidx, myM, 32);
  float* const myOut = out + (long)myRow * DIM;
  const int rSel  = lane & 7;
  const int nl    = lane & 15;
  const int mbase = hiA * 8;

  float acc[8];
#pragma unroll
  for (int r = 0; r < 8; ++r) acc[r] = 0.f;

  const float2* __restrict__ bp = (const float2*)wpack + lane;

  for (int nt = 0; nt < NTILES; ++nt) {
    const float2* bq = bp + (long)nt * (KSTEPS * 32);
    if (nt + 1 < NTILES) __builtin_prefetch(bq + KSTEPS * 32, 0, 3);

    // Two independent accumulator chains -> consecutive WMMAs can pipeline
    v8f c0 = {0.f, 0.f, 0.f, 0.f, 0.f, 0.f, 0.f, 0.f};
    v8f c1 = {0.f, 0.f, 0.f, 0.f, 0.f, 0.f, 0.f, 0.f};
#pragma unroll
    for (int ks = 0; ks < KSTEPS; ks += 2) {
      float2 w0 = bq[(ks + 0) * 32];
      float2 w1 = bq[(ks + 1) * 32];
      v2f b0; b0.x = w0.x; b0.y = w0.y;
      v2f b1v; b1v.x = w1.x; b1v.y = w1.y;
      c0 = __builtin_amdgcn_wmma_f32_16x16x4_f32(false, a[ks + 0], false, b0,
                                                 (short)0, c0, false, false);
      c1 = __builtin_amdgcn_wmma_f32_16x16x4_f32(false, a[ks + 1], false, b1v,
                                                 (short)0, c1, false, false);
    }

    // Fold x: C/D layout n=lane%16, m=r+8*(lane/16). Tile nt covers output
    // row i = nt>>2, x columns j0..j0+15 with j0 = (nt&3)*16.
    const int j0 = (nt & 3) * 16;
#pragma unroll
    for (int r = 0; r < 8; ++r)
      acc[r] = fmaf(c0[r] + c1[r], xt[wave][mbase + r][j0 + nl], acc[r]);

    if ((nt & 3) == 3) {  // finished all 64 j for this i -> reduce over n-lanes
#pragma unroll
      for (int r = 0; r < 8; ++r) {
        float v = acc[r];
        v += __shfl_xor(v, 1, 32);
        v += __shfl_xor(v, 2, 32);
        v += __shfl_xor(v, 4, 32);
        v += __shfl_xor(v, 8, 32);
        acc[r] = v;
      }
      float v = acc[0];
#pragma unroll
      for (int r = 1; r < 8; ++r) v = (rSel == r) ? acc[r] : v;
      if (wr) unsafeAtomicAdd(myOut + (nt >> 2), v);  // global_atomic_add_f32
#pragma unroll
      for (int r = 0; r < 8; ++r) acc[r] = 0.f;
    }
  }
}

extern "C" void kernel_launch(void* const* d_in, const int* in_sizes, int n_in,
                              void* d_out, int out_size, void* d_ws, size_t ws_size,
                              hipStream_t stream) {
  (void)n_in; (void)out_size; (void)ws_size;
  const float* u_emb   = (const float*)d_in[0];
  const float* i_emb   = (const float*)d_in[1];
  const int*   u_of_e  = (const int*)d_in[2];   // user_per_trans
  const int*   i_of_e  = (const int*)d_in[3];   // item_per_trans
  const float* edges_t = (const float*)d_in[4];
  const float* u_t     = (const float*)d_in[5];
  const float* i_t     = (const float*)d_in[6];
  // Param dicts flatten as JAX pytree leaves (keys sorted):
  // W1, W2, W3, b1, b2, b3, be1, be2, g1, g2
  const float* uW1 = (const float*)d_in[7];
  const float* uW2 = (const float*)d_in[8];
  const float* uW3 = (const float*)d_in[9];
  const float* ub1 = (const float*)d_in[10];
  const float* ub2 = (const float*)d_in[11];
  const float* ub3 = (const float*)d_in[12];
  const float* ube1 = (const float*)d_in[13];
  const float* ube2 = (const float*)d_in[14];
  const float* ug1 = (const float*)d_in[15];
  const float* ug2 = (const float*)d_in[16];
  const float* iW1 = (const float*)d_in[17];
  const float* iW2 = (const float*)d_in[18];
  const float* iW3 = (const float*)d_in[19];
  const float* ib1 = (const float*)d_in[20];
  const float* ib2 = (const float*)d_in[21];
  const float* ib3 = (const float*)d_in[22];
  const float* ibe1 = (const float*)d_in[23];
  const float* ibe2 = (const float*)d_in[24];
  const float* ig1 = (const float*)d_in[25];
  const float* ig2 = (const float*)d_in[26];

  const int E  = in_sizes[2];
  const int NU = in_sizes[5];
  const int NI = in_sizes[6];

  float* hLu = (float*)d_out;              // (NU, 64)
  float* hLi = hLu + (long)NU * DIM;       // (NI, 64)
  float* wpI = (float*)d_ws;               // packed W3aug for w_items (1 MB)
  float* wpU = wpI + (long)KPAD * DIM * DIM;

  const int nOut = (NU + NI) * DIM;
  zero_f32<<<(nOut + 255) / 256, 256, 0, stream>>>((float*)d_out, nOut);

  const int packN = NTILES * KSTEPS * 32;
  pack_w3<<<(packN + 255) / 256, 256, 0, stream>>>(iW3, ib3, wpI);
  pack_w3<<<(packN + 255) / 256, 256, 0, stream>>>(uW3, ub3, wpU);

  const int nGroups = (E + 15) / 16;
  const int blocks  = (nGroups + WPB - 1) / WPB;

  // item messages -> hLu: t = u_t[user]-edge_t, net = w_items, x = i_emb[item], scatter by user
  ckconv_side<<<blocks, BLOCK, 0, stream>>>(
      u_t, u_of_e, edges_t, i_emb, i_of_e, u_of_e,
      iW1, ib1, ig1, ibe1, iW2, ib2, ig2, ibe2, wpI, hLu, E);
  // user messages -> hLi: t = i_t[item]-edge_t, net = w_users, x = u_emb[user], scatter by item
  ckconv_side<<<blocks, BLOCK, 0, stream>>>(
      i_t, i_of_e, edges_t, u_emb, u_of_e, i_of_e,
      uW1, ub1, ug1, ube1, uW2, ub2, ug2, ube2, wpU, hLi, E);
}